// BeatPooling_29618094473978
// MI455X (gfx1250) — compile-verified
//
#include <hip/hip_runtime.h>
#include <math.h>

// Problem constants (from reference: B=8, T=8192, D=512, M=512, POS_DIM=32)
#define BB 8
#define TT 8192
#define DD 512
#define MM 512
#define POS_DIM 32
#define KDIM (DD + POS_DIM)   // 544
#define BS 32                 // frames per partial-sum block
#define NB (TT / BS)          // 256 blocks along T

typedef __attribute__((ext_vector_type(2))) float v2f;
typedef __attribute__((ext_vector_type(8))) float v8f;

// ---------------------------------------------------------------------------
// Kernel 1: per-block partial sums.  E[b][blk+1][d] = sum_{t in block} fe[b,t,d]
// grid (NB, B, 2), block 256  -> fully coalesced 1KB reads per iteration.
__global__ void bp_partial_sums(const float* __restrict__ fe, float* __restrict__ E) {
    const int blk = blockIdx.x;
    const int b   = blockIdx.y;
    const int d   = blockIdx.z * 256 + threadIdx.x;
    const float* p = fe + ((size_t)(b * TT + blk * BS)) * DD + d;
    float acc = 0.0f;
#pragma unroll
    for (int i = 0; i < BS; ++i) acc += p[(size_t)i * DD];
    E[((size_t)(b * (NB + 1) + blk + 1)) * DD + d] = acc;
}

// ---------------------------------------------------------------------------
// Kernel 2: serial scan over blocks per (b,d) column, in place:
// afterwards E[b][k][d] = sum of all blocks < k  (exclusive block prefix).
// B*D = 4096 threads, 256 coalesced iterations each.
__global__ void bp_scan_blocks(float* __restrict__ E) {
    const int idx = blockIdx.x * blockDim.x + threadIdx.x;   // 0..B*D-1
    const int b = idx / DD;
    const int d = idx % DD;
    float* base = E + (size_t)b * (NB + 1) * DD + d;
    float run = 0.0f;
    base[0] = 0.0f;
    for (int k = 1; k <= NB; ++k) {
        run += base[(size_t)k * DD];
        base[(size_t)k * DD] = run;
    }
}

// ---------------------------------------------------------------------------
// Kernel 3: per-segment pooling + Fourier features -> A[b*M+m][0..543]
// sum[s,e) = E[eb] - E[sb] + sum[eb*BS, e) - sum[sb*BS, s)
// grid (M, B), block 256 (each thread owns channels d and d+256).
__global__ void bp_pool_fourier(const float* __restrict__ fe,
                                const int*   __restrict__ bounds,
                                const float* __restrict__ E,
                                float*       __restrict__ A) {
    const int m = blockIdx.x;
    const int b = blockIdx.y;
    const int tid = threadIdx.x;

    int s = bounds[(b * MM + m) * 2 + 0];
    int e = bounds[(b * MM + m) * 2 + 1];
    s = min(max(s, 0), TT - 1);
    e = max(s + 1, min(e, TT));
    const int sb = s / BS;
    const int eb = e / BS;             // eb may be NB when e == T (E has NB+1 rows)
    const float inv = 1.0f / (float)(e - s);

    const float* Eb  = E  + (size_t)b * (NB + 1) * DD;
    const float* feb = fe + (size_t)b * TT * DD;
    float* Arow = A + (size_t)(b * MM + m) * KDIM;

#pragma unroll
    for (int h = 0; h < 2; ++h) {
        const int d = tid + h * 256;
        float acc = Eb[(size_t)eb * DD + d] - Eb[(size_t)sb * DD + d];
        for (int t = eb * BS; t < e; ++t) acc += feb[(size_t)t * DD + d];  // + tail_e
        for (int t = sb * BS; t < s; ++t) acc -= feb[(size_t)t * DD + d];  // - tail_s
        Arow[d] = acc * inv;
    }

    if (tid < POS_DIM) {
        const float pos  = (float)m / (float)(MM - 1);
        const int   j    = tid & 15;                       // half = 16
        const float freq = expf((float)j * (logf(1000.0f) / 15.0f));
        const float ang  = pos * freq;
        Arow[DD + tid] = (tid < 16) ? sinf(ang) : cosf(ang);
    }
}

// ---------------------------------------------------------------------------
// Kernel 4: C[4096x512] = A[4096x544] @ W[544x512] + bias, full-f32 WMMA.
// One wave per 16x16 output tile, K stepped by 4 via V_WMMA_F32_16X16X4_F32.
//
// Lane layouts per CDNA5 ISA §7.12.2 (32-bit operands):
//   A 16x4 : lane&15 = M row; half-wave selects K pair {0,1} vs {2,3} in v[0],v[1]
//   B 4x16 : lane&15 = N col; half-wave selects K row pair; v[0]=K, v[1]=K+1
//   C 16x16: VGPR i -> row i (lanes 0-15) / row i+8 (lanes 16-31), N = lane&15
__global__ void bp_gemm_wmma(const float* __restrict__ A,
                             const float* __restrict__ W,
                             const float* __restrict__ bias,
                             float*       __restrict__ out) {
    const int wave = threadIdx.x >> 5;                 // 8 waves per block
    const int lane = threadIdx.x & 31;
    const int colBase = blockIdx.x * 16;
    const int rowBase = (blockIdx.y * 8 + wave) * 16;
    const int lr = lane & 15;
    const int kh = (lane >> 4) * 2;                    // 0 or 2

    const float* Ap = A + (size_t)(rowBase + lr) * KDIM + kh;
    const float* Wp = W + (size_t)kh * DD + colBase + lr;

    v8f acc = {};
#pragma unroll 4
    for (int k0 = 0; k0 < KDIM; k0 += 4) {
        v2f av, bv;
        av.x = Ap[k0];
        av.y = Ap[k0 + 1];
        bv.x = Wp[(size_t)k0 * DD];
        bv.y = Wp[(size_t)(k0 + 1) * DD];
        acc = __builtin_amdgcn_wmma_f32_16x16x4_f32(
            /*neg_a=*/false, av, /*neg_b=*/false, bv,
            /*c_mod=*/(short)0, acc, /*reuse_a=*/false, /*reuse_b=*/false);
    }

    const float bvv = bias[colBase + lr];
    const int row0 = rowBase + (lane >> 4) * 8;
#pragma unroll
    for (int i = 0; i < 8; ++i) {
        out[(size_t)(row0 + i) * DD + colBase + lr] = acc[i] + bvv;
    }
}

// ---------------------------------------------------------------------------
extern "C" void kernel_launch(void* const* d_in, const int* in_sizes, int n_in,
                              void* d_out, int out_size, void* d_ws, size_t ws_size,
                              hipStream_t stream) {
    const float* fe     = (const float*)d_in[0];   // [B,T,D] f32
    const int*   bounds = (const int*)  d_in[1];   // [B,M,2] int32
    const float* W      = (const float*)d_in[2];   // [544,512] f32
    const float* bias   = (const float*)d_in[3];   // [512] f32
    float* out = (float*)d_out;                    // [B,M,D] f32

    // Workspace carve-up (~12.5 MB total):
    float* E = (float*)d_ws;                           // B*(NB+1)*D floats (4.2 MB)
    float* A = E + (size_t)BB * (NB + 1) * DD;         // B*M*KDIM floats  (8.9 MB)

    bp_partial_sums<<<dim3(NB, BB, 2), 256, 0, stream>>>(fe, E);
    bp_scan_blocks <<<(BB * DD) / 256, 256, 0, stream>>>(E);
    bp_pool_fourier<<<dim3(MM, BB), 256, 0, stream>>>(fe, bounds, E, A);
    bp_gemm_wmma   <<<dim3(DD / 16, (BB * MM) / 128), 256, 0, stream>>>(A, W, bias, out);
}